// SpatialDyadicRelationalGC_54649163875005
// MI455X (gfx1250) — compile-verified
//
#include <hip/hip_runtime.h>

typedef __attribute__((ext_vector_type(2))) float v2f;
typedef __attribute__((ext_vector_type(8))) float v8f;

#define Nn    32
#define CIN   64
#define Tt    300
#define Vv    25
#define K3    3
#define COUT  64
#define P     32   // padded pixel (v) dimension, 2 WMMA N-tiles

// D = A(16x4 f32) * B(4x16 f32) + C(16x16 f32)
__device__ __forceinline__ v8f wmma4(v2f a, v2f b, v8f c) {
    return __builtin_amdgcn_wmma_f32_16x16x4_f32(
        /*neg_a=*/false, a, /*neg_b=*/false, b,
        /*c_mod=*/(short)0, c, /*reuse_a=*/false, /*reuse_b=*/false);
}

__global__ __launch_bounds__(256)
void sdrgc_fused(const float* __restrict__ x,   const float* __restrict__ RAM,
                 const float* __restrict__ A,   const float* __restrict__ Wc,
                 const float* __restrict__ bc,  const float* __restrict__ Wr,
                 const float* __restrict__ br,  float* __restrict__ out)
{
    __shared__ float xs[CIN * P];        //  8192 B : x tile (one stream at a time)
    __shared__ float spre[192 * P];      // 24576 B : conv output (192 ch, one stream)
    __shared__ float r_s[2][COUT * P];   // 16384 B : r_a, r_b (kept for both streams)
    __shared__ float A_s[K3 * Vv * Vv];  //  7500 B
    __shared__ float RAMn[Vv * Vv];      //  2500 B : RAM[v][w]
    __shared__ float RAMt[Vv * Vv];      //  2500 B : RAM[v][w] stored at [w][v]

    const int wg   = blockIdx.x;
    const int n    = wg / Tt;
    const int t    = wg % Tt;
    const int tid  = threadIdx.x;
    const int wave = tid >> 5;
    const int lane = tid & 31;
    const int lo   = lane & 15;
    const int hi   = lane >> 4;

    // ---------------- stage small constants + x(stream a) ----------------
    for (int i = tid; i < K3 * Vv * Vv; i += 256) A_s[i] = A[i];
    for (int i = tid; i < Vv * Vv; i += 256) {
        float val = RAM[(size_t)n * Tt * Vv * Vv + (size_t)t * Vv * Vv + i];
        int v = i / Vv, w = i % Vv;
        RAMn[i] = val;
        RAMt[w * Vv + v] = val;
    }
    const size_t xbaseA = (size_t)(n * 2 + 0) * CIN * Tt * Vv + (size_t)t * Vv;
    const size_t xbaseB = (size_t)(n * 2 + 1) * CIN * Tt * Vv + (size_t)t * Vv;
    for (int i = tid; i < CIN * P; i += 256) {
        int c = i >> 5, p = i & 31;
        xs[i] = (p < Vv) ? x[xbaseA + (size_t)c * Tt * Vv + p] : 0.0f;
    }

    // ---------------- preload W operands (A-matrix fragments) -------------
    // wave owns output-channel tiles {2*wave, 2*wave+1} of 16 rows each.
    // A-operand layout: lane -> row = lo, VGPR j with half 'hi' -> k = 2*hi + j.
    v2f wop[2][16];
    #pragma unroll
    for (int mm = 0; mm < 2; ++mm) {
        const int mt  = 2 * wave + mm;
        const int row = mt * 16 + lo;
        const float* Wp = (mt < 12) ? (Wc + row * CIN) : (Wr + (row - 192) * CIN);
        #pragma unroll
        for (int kk = 0; kk < 16; ++kk) {
            wop[mm][kk] = *(const v2f*)(Wp + kk * 4 + 2 * hi);
        }
    }

    // conv GEMM for one stream: [256 x 64] * [64 x 32] -> spre(192 rows) + rdst(64 rows)
    auto conv_stream = [&](float* rdst) {
        v8f acc[2][2];
        #pragma unroll
        for (int mm = 0; mm < 2; ++mm)
            #pragma unroll
            for (int nt = 0; nt < 2; ++nt)
                acc[mm][nt] = (v8f)(0.0f);
        #pragma unroll
        for (int kk = 0; kk < 16; ++kk) {
            v2f bop[2];
            #pragma unroll
            for (int nt = 0; nt < 2; ++nt) {
                const int col = nt * 16 + lo;
                const int r0  = kk * 4 + 2 * hi;   // B row = 2*hi + j
                bop[nt].x = xs[(r0 + 0) * P + col];
                bop[nt].y = xs[(r0 + 1) * P + col];
            }
            #pragma unroll
            for (int mm = 0; mm < 2; ++mm)
                #pragma unroll
                for (int nt = 0; nt < 2; ++nt)
                    acc[mm][nt] = wmma4(wop[mm][kk], bop[nt], acc[mm][nt]);
        }
        // C layout: VGPR rr -> row = rr + 8*hi, col = lo
        #pragma unroll
        for (int mm = 0; mm < 2; ++mm) {
            const int mt = 2 * wave + mm;
            #pragma unroll
            for (int nt = 0; nt < 2; ++nt) {
                const int col = nt * 16 + lo;
                #pragma unroll
                for (int rr = 0; rr < 8; ++rr) {
                    const int row  = mt * 16 + rr + 8 * hi;
                    const float bb = (row < 192) ? bc[row] : br[row - 192];
                    const float vv = acc[mm][nt][rr] + bb;
                    if (row < 192) spre[row * P + col] = vv;
                    else           rdst[(row - 192) * P + col] = vv;
                }
            }
        }
    };

    // adjacency GEMM: C[c,w] += sum_{k3,v} spre[k3*64+c, v] * A[k3,v,w]
    // wave -> c-tile (wave>>1), w-tile (wave&1)
    auto phaseA = [&](v8f& C) {
        const int c    = (wave >> 1) * 16 + lo;
        const int wcol = (wave & 1) * 16 + lo;
        #pragma unroll
        for (int k3 = 0; k3 < K3; ++k3) {
            #pragma unroll
            for (int vs = 0; vs < 7; ++vs) {
                const int v0 = vs * 4 + 2 * hi;
                v2f a, b;
                a.x = spre[(k3 * 64 + c) * P + v0];
                a.y = spre[(k3 * 64 + c) * P + v0 + 1];
                b.x = (v0     < Vv && wcol < Vv) ? A_s[k3 * Vv * Vv + (v0    ) * Vv + wcol] : 0.0f;
                b.y = (v0 + 1 < Vv && wcol < Vv) ? A_s[k3 * Vv * Vv + (v0 + 1) * Vv + wcol] : 0.0f;
                C = wmma4(a, b, C);
            }
        }
    };

    // relational GEMM: C[c,ncol] += sum_w rsrc[c,w] * Rm[w,ncol]   (Rm 25x25, zero-padded via checks)
    auto phaseR = [&](v8f& C, const float* rsrc, const float* Rm) {
        const int c    = (wave >> 1) * 16 + lo;
        const int ncol = (wave & 1) * 16 + lo;
        #pragma unroll
        for (int ws = 0; ws < 7; ++ws) {
            const int w0 = ws * 4 + 2 * hi;
            v2f a, b;
            a.x = rsrc[c * P + w0];
            a.y = rsrc[c * P + w0 + 1];
            b.x = (w0     < Vv && ncol < Vv) ? Rm[(w0    ) * Vv + ncol] : 0.0f;
            b.y = (w0 + 1 < Vv && ncol < Vv) ? Rm[(w0 + 1) * Vv + ncol] : 0.0f;
            C = wmma4(a, b, C);
        }
    };

    auto store_out = [&](const v8f& C, int s) {
        const int ncol = (wave & 1) * 16 + lo;
        if (ncol < Vv) {
            #pragma unroll
            for (int rr = 0; rr < 8; ++rr) {
                const int c = (wave >> 1) * 16 + rr + 8 * hi;
                out[((size_t)(n * 2 + s) * COUT + c) * Tt * Vv + (size_t)t * Vv + ncol] = C[rr];
            }
        }
    };

    __syncthreads();                 // staging (A_s, RAM, xa) complete
    conv_stream(r_s[0]);             // spre <- stream a (192 ch), r_a
    __syncthreads();                 // spre/r_a visible; xs reads finished

    // overlap: refill xs with stream b while consuming spre(a)
    for (int i = tid; i < CIN * P; i += 256) {
        int c = i >> 5, p = i & 31;
        xs[i] = (p < Vv) ? x[xbaseB + (size_t)c * Tt * Vv + p] : 0.0f;
    }
    v8f Ca = (v8f)(0.0f);
    v8f Cb = (v8f)(0.0f);
    phaseA(Ca);                      // sa
    __syncthreads();                 // xs(b) ready, spre(a) reads done
    conv_stream(r_s[1]);             // spre <- stream b, r_b
    __syncthreads();
    phaseA(Cb);                      // sb
    phaseR(Ca, r_s[1], RAMt);        // a_dr = r_b * RAM^T
    phaseR(Cb, r_s[0], RAMn);        // b_dr = r_a * RAM
    store_out(Ca, 0);
    store_out(Cb, 1);
}

extern "C" void kernel_launch(void* const* d_in, const int* in_sizes, int n_in,
                              void* d_out, int out_size, void* d_ws, size_t ws_size,
                              hipStream_t stream) {
    const float* x   = (const float*)d_in[0];
    const float* RAM = (const float*)d_in[1];
    const float* A   = (const float*)d_in[2];
    const float* Wc  = (const float*)d_in[3];
    const float* bc  = (const float*)d_in[4];
    const float* Wr  = (const float*)d_in[5];
    const float* br  = (const float*)d_in[6];
    float* out = (float*)d_out;

    sdrgc_fused<<<dim3(Nn * Tt), dim3(256), 0, stream>>>(x, RAM, A, Wc, bc, Wr, br, out);

    // reference returns (out, RAM, A): append RAM and A to d_out
    const size_t outElems = (size_t)Nn * 2 * COUT * Tt * Vv;   // 30,720,000
    const size_t ramElems = (size_t)Nn * Tt * Vv * Vv;         //  6,000,000
    hipMemcpyAsync(out + outElems, RAM, ramElems * sizeof(float),
                   hipMemcpyDeviceToDevice, stream);
    hipMemcpyAsync(out + outElems + ramElems, A, (size_t)K3 * Vv * Vv * sizeof(float),
                   hipMemcpyDeviceToDevice, stream);
}